// ChebNet_7576322310704
// MI455X (gfx1250) — compile-verified
//
#include <hip/hip_runtime.h>

typedef __attribute__((ext_vector_type(2))) float v2f;
typedef __attribute__((ext_vector_type(8))) float v8f;

// ---------------- graph norm ----------------
__global__ void cheb_deg_kernel(const int* __restrict__ ei, float* __restrict__ deg, int E_) {
    int e = blockIdx.x * blockDim.x + threadIdx.x;
    if (e < E_) atomicAdd(&deg[ei[e]], 1.0f);          // segment_sum of ones over src
}

__global__ void cheb_dinv_kernel(float* __restrict__ deg, int N_) {
    int i = blockIdx.x * blockDim.x + threadIdx.x;
    if (i < N_) {
        float d = deg[i];
        deg[i] = (d > 0.0f) ? rsqrtf(d) : 0.0f;        // max(d,1)==d when d>=1
    }
}

__global__ void cheb_norm_kernel(const int* __restrict__ ei, const float* __restrict__ dinv,
                                 float* __restrict__ nrm, int E_) {
    int e = blockIdx.x * blockDim.x + threadIdx.x;
    if (e < E_) nrm[e] = -dinv[ei[e]] * dinv[ei[E_ + e]];
}

// ---------------- scalar propagate (layer-1 features, dim 1) ----------------
__global__ void cheb_prop1_kernel(const int* __restrict__ ei, const float* __restrict__ nrm,
                                  const float* __restrict__ xin, float* __restrict__ yout, int E_) {
    int e = blockIdx.x * blockDim.x + threadIdx.x;
    if (e < E_) atomicAdd(&yout[ei[E_ + e]], nrm[e] * xin[ei[e]]);
}

// ---------------- layer 1 fused (1 -> 32, K=3) ----------------
__global__ void cheb_layer1_kernel(const float* __restrict__ x, const float* __restrict__ t1,
                                   const float* __restrict__ t2raw, const float* __restrict__ W1,
                                   const float* __restrict__ b1, float* __restrict__ h, int N_) {
    int t = blockIdx.x * blockDim.x + threadIdx.x;
    int i = t >> 5, j = t & 31;
    if (i < N_) {
        float a = x[i];
        float b = t1[i];
        float c = 2.0f * t2raw[i] - a;                 // Tx2 = 2*prop(Tx1) - Tx0
        float v = a * W1[j] + b * W1[32 + j] + c * W1[64 + j] + b1[j];
        h[(size_t)i * 32 + j] = fmaxf(v, 0.0f);
    }
}

// ---------------- 32-feature propagate: one wave per edge ----------------
__global__ void cheb_prop32_kernel(const int* __restrict__ ei, const float* __restrict__ nrm,
                                   const float* __restrict__ Xin, float* __restrict__ Yout, int E_) {
    long long t = (long long)blockIdx.x * blockDim.x + threadIdx.x;
    int e = (int)(t >> 5);
    int j = (int)(t & 31);
    if (e < E_) {
        int s = ei[e];
        int d = ei[E_ + e];
        float v = nrm[e] * Xin[(size_t)s * 32 + j];    // 128B coalesced gather (L2-resident)
        atomicAdd(&Yout[(size_t)d * 32 + j], v);       // 128B coalesced f32 atomic scatter
    }
}

// ---------------- effective weights: WA=W2[0]-W2[2], WB=W2[1], WC=2*W2[2] ----------------
__global__ void cheb_wprep_kernel(const float* __restrict__ W2, float* __restrict__ Wx) {
    int t = blockIdx.x * blockDim.x + threadIdx.x;
    if (t < 1024) {
        Wx[t]        = W2[t] - W2[2048 + t];
        Wx[1024 + t] = W2[1024 + t];
        Wx[2048 + t] = 2.0f * W2[2048 + t];
    }
}

// ---------------- layer 2 + final FC, fused, WMMA f32 16x16x4 ----------------
// One wave = 16 nodes x 32 features. D = Tx0@WA + P1@WB + P2@WC + b2, relu, then @Wfc + bfc.
__global__ __launch_bounds__(256) void cheb_l2fc_wmma_kernel(
    const float* __restrict__ Tx0, const float* __restrict__ P1v, const float* __restrict__ P2v,
    const float* __restrict__ Wx, const float* __restrict__ b2,
    const float* __restrict__ Wfc, const float* __restrict__ bfc,
    float* __restrict__ out, int N_, int nTiles) {
    int wave = (blockIdx.x * blockDim.x + threadIdx.x) >> 5;
    int lane = threadIdx.x & 31;
    if (wave >= nTiles) return;                        // whole-wave exit; EXEC all-1s for WMMA

    int m0   = wave * 16;
    int ln   = lane & 15;
    int half = lane >> 4;

    // seed accumulators with the bias (C/D layout: VGPR v, lanes0-15 -> M=v, lanes16-31 -> M=v+8)
    v8f c0, c1;
    float bz0 = b2[ln], bz1 = b2[16 + ln];
#pragma unroll
    for (int v = 0; v < 8; ++v) { c0[v] = bz0; c1[v] = bz1; }

    int arow = m0 + ln; if (arow >= N_) arow = N_ - 1; // tail guard (loads only)
    const float* mats[3] = {Tx0, P1v, P2v};
#pragma unroll
    for (int mi = 0; mi < 3; ++mi) {
        const float* X = mats[mi] + (size_t)arow * 32 + half * 2;  // A: K={k0,k0+1} lo-half, {k0+2,k0+3} hi-half
        const float* W = Wx + mi * 1024;                            // 32x32 row-major [k][n]
#pragma unroll
        for (int k0 = 0; k0 < 32; k0 += 4) {
            v2f a;  a.x  = X[k0];                      a.y  = X[k0 + 1];
            v2f b0; b0.x = W[(k0 + half * 2) * 32 + ln];        b0.y = W[(k0 + half * 2 + 1) * 32 + ln];
            v2f b1; b1.x = W[(k0 + half * 2) * 32 + 16 + ln];   b1.y = W[(k0 + half * 2 + 1) * 32 + 16 + ln];
            c0 = __builtin_amdgcn_wmma_f32_16x16x4_f32(false, a, false, b0, (short)0, c0, false, false);
            c1 = __builtin_amdgcn_wmma_f32_16x16x4_f32(false, a, false, b1, (short)0, c1, false, false);
        }
    }

    // epilogue: relu, multiply by Wfc, butterfly-reduce 16 lanes per half, + bfc
    float wl0 = Wfc[ln], wl1 = Wfc[16 + ln];
    float bias = bfc[0];
#pragma unroll
    for (int v = 0; v < 8; ++v) {
        float s = fmaxf(c0[v], 0.0f) * wl0 + fmaxf(c1[v], 0.0f) * wl1;
        s += __shfl_xor(s, 1, 32);
        s += __shfl_xor(s, 2, 32);
        s += __shfl_xor(s, 4, 32);
        s += __shfl_xor(s, 8, 32);
        int row = m0 + v + half * 8;                   // lane 0 -> M=v, lane 16 -> M=v+8
        if (ln == 0 && row < N_) out[row] = s + bias;
    }
}

extern "C" void kernel_launch(void* const* d_in, const int* in_sizes, int n_in,
                              void* d_out, int out_size, void* d_ws, size_t ws_size,
                              hipStream_t stream) {
    const float* x   = (const float*)d_in[0];
    const int*   ei  = (const int*)d_in[1];   // (2,E): [0..E)=src, [E..2E)=dst
    const float* W1  = (const float*)d_in[2];
    const float* b1  = (const float*)d_in[3];
    const float* W2  = (const float*)d_in[4];
    const float* b2  = (const float*)d_in[5];
    const float* Wfc = (const float*)d_in[6];
    const float* bfc = (const float*)d_in[7];
    float* out = (float*)d_out;

    const int N_ = in_sizes[0];               // 100000
    const int E_ = in_sizes[1] / 2;           // 2000000

    // workspace layout (floats)
    float* deg = (float*)d_ws;                // N  (becomes dinv in place)
    float* nrm = deg + N_;                    // E
    float* t1  = nrm + E_;                    // N
    float* t2  = t1 + N_;                     // N
    float* h   = t2 + N_;                     // N*32
    float* P1  = h  + (size_t)N_ * 32;        // N*32
    float* P2  = P1 + (size_t)N_ * 32;        // N*32
    float* Wx  = P2 + (size_t)N_ * 32;        // 3*1024

    const int T = 256;
    int gE  = (E_ + T - 1) / T;
    int gN  = (N_ + T - 1) / T;
    int gN32 = (N_ * 32 + T - 1) / T;
    long long e32 = (long long)E_ * 32;
    int gE32 = (int)((e32 + T - 1) / T);

    // norm
    hipMemsetAsync(deg, 0, (size_t)N_ * sizeof(float), stream);
    cheb_deg_kernel<<<gE, T, 0, stream>>>(ei, deg, E_);
    cheb_dinv_kernel<<<gN, T, 0, stream>>>(deg, N_);
    cheb_norm_kernel<<<gE, T, 0, stream>>>(ei, deg, nrm, E_);

    // layer 1 (scalar features)
    hipMemsetAsync(t1, 0, (size_t)N_ * sizeof(float), stream);
    cheb_prop1_kernel<<<gE, T, 0, stream>>>(ei, nrm, x, t1, E_);
    hipMemsetAsync(t2, 0, (size_t)N_ * sizeof(float), stream);
    cheb_prop1_kernel<<<gE, T, 0, stream>>>(ei, nrm, t1, t2, E_);
    cheb_layer1_kernel<<<gN32, T, 0, stream>>>(x, t1, t2, W1, b1, h, N_);

    // layer 2 propagates (32 features)
    hipMemsetAsync(P1, 0, (size_t)N_ * 32 * sizeof(float), stream);
    cheb_prop32_kernel<<<gE32, T, 0, stream>>>(ei, nrm, h, P1, E_);
    hipMemsetAsync(P2, 0, (size_t)N_ * 32 * sizeof(float), stream);
    cheb_prop32_kernel<<<gE32, T, 0, stream>>>(ei, nrm, P1, P2, E_);

    // effective weights + fused GEMM/FC epilogue (WMMA)
    cheb_wprep_kernel<<<4, T, 0, stream>>>(W2, Wx);
    int nTiles = (N_ + 15) / 16;
    int gW = (nTiles + 7) / 8;                // 8 waves per 256-thread block
    cheb_l2fc_wmma_kernel<<<gW, T, 0, stream>>>(h, P1, P2, Wx, b2, Wfc, bfc, out, N_, nTiles);
}